// SingleHeadAttention_87960930222113
// MI455X (gfx1250) — compile-verified
//
#include <hip/hip_runtime.h>

typedef __attribute__((ext_vector_type(2))) float v2f;
typedef __attribute__((ext_vector_type(4))) float v4f;
typedef __attribute__((ext_vector_type(8))) float v8f;
typedef __attribute__((ext_vector_type(4))) unsigned tdm_g0_t;   // D# group 0
typedef __attribute__((ext_vector_type(8))) int      tdm_g1_t;   // D# group 1
typedef __attribute__((ext_vector_type(4))) int      tdm_g2_t;   // D# groups 2/3
typedef __attribute__((ext_vector_type(8))) int      tdm_g1b_t;  // 6-arg variant extra

#define SEQ  4096
#define EMB  1024
#define AD   128
#define NBAT 4

#if defined(__has_builtin)
#  if __has_builtin(__builtin_amdgcn_tensor_load_to_lds) && \
      __has_builtin(__builtin_amdgcn_s_wait_tensorcnt)
#    define USE_TDM 1
#  endif
#endif
#ifndef USE_TDM
#  define USE_TDM 0
#endif

__device__ __forceinline__ v8f wmma_f32(v2f a, v2f b, v8f c) {
    // (neg_a, A, neg_b, B, c_mod, C, reuse_a, reuse_b)
    return __builtin_amdgcn_wmma_f32_16x16x4_f32(false, a, false, b, (short)0, c,
                                                 false, false);
}

#if USE_TDM
__device__ __forceinline__ unsigned lds_off(const void* p) {
    // flat LDS addr = aperture(high 32b) | byte offset(low 32b)  (ISA 10.2)
    return (unsigned)(unsigned long long)p;
}

// 2-D tile load via Tensor Data Mover.  Row-major f32 tensor:
//   tile = tile_d1 rows x tile_d0 elems, row stride = stride_e elems.
// HW inserts LDS padding: (pad_amt_code+1) DWORDs after every 2<<pad_int_code DWORDs.
__device__ __forceinline__ void tdm_load_2d(const void* gptr, unsigned ldsa,
                                            unsigned tile_d0, unsigned tile_d1,
                                            unsigned long long stride_e,
                                            unsigned tensor_d0, unsigned tensor_d1,
                                            unsigned pad_int_code, unsigned pad_amt_code)
{
    const unsigned long long ga = (unsigned long long)gptr;
    tdm_g0_t g0;
    g0[0] = 1u;                                       // count=1, user descriptor
    g0[1] = ldsa;                                     // lds_addr [63:32]
    g0[2] = (unsigned)ga;                             // global_addr [95:64]
    g0[3] = (unsigned)((ga >> 32) & 0x01FFFFFFu)      // global_addr [120:96]
          | (2u << 30);                               // type = 2 ("image")
    tdm_g1_t g1;
    g1[0] = (int)((2u << 16)                          // data_size = 4B
          | (1u << 20)                                // pad_enable
          | (pad_int_code << 22) | (pad_amt_code << 25));
    g1[1] = (int)((tensor_d0 & 0xFFFFu) << 16);                       // td0 lo
    g1[2] = (int)((tensor_d0 >> 16) | ((tensor_d1 & 0xFFFFu) << 16)); // td0 hi, td1 lo
    g1[3] = (int)((tensor_d1 >> 16) | (tile_d0 << 16));               // td1 hi, tile0
    g1[4] = (int)tile_d1;                                             // tile1 (tile2=0)
    g1[5] = (int)(unsigned)(stride_e & 0xFFFFFFFFu);                  // dim0 stride lo
    g1[6] = (int)(unsigned)((stride_e >> 32) & 0xFFFFu);              // dim0 stride hi
    g1[7] = 0;                                                        // dim1 stride (2D)
    tdm_g2_t z4 = {};
#  if __has_include(<hip/amd_detail/amd_gfx1250_TDM.h>)
    tdm_g1b_t z8 = {};
    __builtin_amdgcn_tensor_load_to_lds(g0, g1, z4, z4, z8, 0);       // clang-23 arity
#  else
    __builtin_amdgcn_tensor_load_to_lds(g0, g1, z4, z4, 0);           // ROCm 7.2 arity
#  endif
}
#endif // USE_TDM

// ---------------------------------------------------------------------------
// Kernel 1: fused QKV projection.  Q/K/V = X @ W^T   (fp32 WMMA 16x16x4)
// Block: 256 threads = 8 waves, owns 16 rows of X.  X chunks [16x256] arrive
// via TDM (double buffered, HW row-padded to stride 260 -> conflict-free LDS).
// Wave w owns output columns [16w,16w+16) for Q, K and V (A-frag reused 3x).
// ---------------------------------------------------------------------------
__global__ __launch_bounds__(256)
void qkv_proj_kernel(const float* __restrict__ X,
                     const float* __restrict__ Wq,
                     const float* __restrict__ Wk,
                     const float* __restrict__ Wv,
                     float* __restrict__ Qo,
                     float* __restrict__ Ko,
                     float* __restrict__ Vo)
{
#if USE_TDM
    __shared__ float Xs[2][16][260];
#else
    __shared__ float Xs[1][16][260];
#endif
    const int tid  = threadIdx.x;
    const int lane = tid & 31;
    const int wv   = tid >> 5;
    const int m0   = blockIdx.x * 16;
    const int n0   = wv * 16;
    const int hi   = lane >> 4;
    const int lm   = lane & 15;

    v8f accq = {}; v8f acck = {}; v8f accv = {};

#if USE_TDM
    if (wv == 0)
        tdm_load_2d(X + (size_t)m0 * EMB, lds_off(&Xs[0][0][0]),
                    256, 16, EMB, EMB, NBAT * SEQ, 7, 3);   // pad 4dw / 256dw
#endif

    for (int c4 = 0; c4 < 4; ++c4) {
#if USE_TDM
        if (wv == 0) {
            if (c4 < 3) {
                tdm_load_2d(X + (size_t)m0 * EMB + (c4 + 1) * 256,
                            lds_off(&Xs[(c4 + 1) & 1][0][0]),
                            256, 16, EMB, EMB, NBAT * SEQ, 7, 3);
                __builtin_amdgcn_s_wait_tensorcnt(1);   // current chunk resident
            } else {
                __builtin_amdgcn_s_wait_tensorcnt(0);
            }
        }
        __syncthreads();
        const float (*Xc)[260] = Xs[c4 & 1];
#else
        {
            const int row  = tid >> 4;
            const int colb = (tid & 15) * 16;
            const float* src = X + (size_t)(m0 + row) * EMB + c4 * 256 + colb;
            #pragma unroll
            for (int u = 0; u < 4; ++u)
                *(v4f*)&Xs[0][row][colb + u * 4] = *(const v4f*)(src + u * 4);
        }
        __syncthreads();
        const float (*Xc)[260] = Xs[0];
#endif

        #pragma unroll 4
        for (int kk = 0; kk < 64; ++kk) {
            const int kb = kk * 4 + 2 * hi;
            v2f a = { Xc[lm][kb], Xc[lm][kb + 1] };     // A frag, shared by Q/K/V

            const int e = c4 * 256 + kb;
            const int n = n0 + lm;
            const float* wqp = Wq + (size_t)n * EMB + e;
            const float* wkp = Wk + (size_t)n * EMB + e;
            const float* wvp = Wv + (size_t)n * EMB + e;
            v2f bq = { wqp[0], wqp[1] };
            v2f bk = { wkp[0], wkp[1] };
            v2f bv = { wvp[0], wvp[1] };

            accq = wmma_f32(a, bq, accq);
            acck = wmma_f32(a, bk, acck);
            accv = wmma_f32(a, bv, accv);
        }
        __syncthreads();
    }

    #pragma unroll
    for (int i = 0; i < 8; ++i) {
        const size_t idx = (size_t)(m0 + hi * 8 + i) * AD + (n0 + lm);
        Qo[idx] = accq[i];
        Ko[idx] = acck[i];
        Vo[idx] = accv[i];
    }
}

// ---------------------------------------------------------------------------
// Kernel 2: causal flash attention, fp32 WMMA.  4 waves/block, each owning a
// 16-query tile; the 4 tiles share TDM-staged (double buffered, HW-padded)
// K/V tiles.  All compute guards are wave-uniform -> EXEC all-1s for WMMA.
// ---------------------------------------------------------------------------
#define AWAVES 4

__global__ __launch_bounds__(32 * AWAVES)
void attn_kernel(const float* __restrict__ Q,
                 const float* __restrict__ K,
                 const float* __restrict__ V,
                 float* __restrict__ O)
{
#if USE_TDM
    __shared__ float Ks[2][16][132];
    __shared__ float Vs[2][16][132];
#else
    __shared__ float Ks[1][16][132];
    __shared__ float Vs[1][16][132];
#endif
    __shared__ float Ps[AWAVES][16][20];

    const int tid   = threadIdx.x;
    const int lane  = tid & 31;
    const int wv    = tid >> 5;
    const int lm    = lane & 15;
    const int hi    = lane >> 4;
    const int b     = blockIdx.x >> 6;              // 64 blocks per batch
    const int qt0   = (blockIdx.x & 63) * AWAVES;
    const int qt    = qt0 + wv;                     // this wave's q-tile
    const int qtmax = qt0 + AWAVES - 1;
    const size_t base = (size_t)b * SEQ * AD;

    // Q A-fragments for the whole tile (K = attn dim 128, 32 steps of 4)
    v2f qa[32];
    {
        const float* qp = Q + base + (size_t)(qt * 16 + lm) * AD + 2 * hi;
        #pragma unroll
        for (int e = 0; e < 32; ++e) {
            qa[e].x = qp[e * 4 + 0];
            qa[e].y = qp[e * 4 + 1];
        }
    }

    v8f o[8];
    #pragma unroll
    for (int t = 0; t < 8; ++t) o[t] = (v8f){};
    float mrow[8], lrow[8];
    #pragma unroll
    for (int i = 0; i < 8; ++i) { mrow[i] = -__builtin_inff(); lrow[i] = 0.0f; }

    const float scale = 0.08838834764831845f;       // 1/sqrt(128)

#if USE_TDM
    if (wv == 0) {
        tdm_load_2d(K + base, lds_off(&Ks[0][0][0]), 128, 16, AD, AD, SEQ, 6, 3);
        tdm_load_2d(V + base, lds_off(&Vs[0][0][0]), 128, 16, AD, AD, SEQ, 6, 3);
    }
#endif

    for (int j = 0; j <= qtmax; ++j) {
#if USE_TDM
        const int cur = j & 1;
        if (wv == 0) {
            if (j < qtmax) {                        // prefetch tile j+1 via TDM
                const float* kn = K + base + (size_t)((j + 1) * 16) * AD;
                const float* vn = V + base + (size_t)((j + 1) * 16) * AD;
                tdm_load_2d(kn, lds_off(&Ks[cur ^ 1][0][0]), 128, 16, AD, AD, SEQ, 6, 3);
                tdm_load_2d(vn, lds_off(&Vs[cur ^ 1][0][0]), 128, 16, AD, AD, SEQ, 6, 3);
                __builtin_amdgcn_s_wait_tensorcnt(2);   // tile j resident, j+1 in flight
            } else {
                __builtin_amdgcn_s_wait_tensorcnt(0);
            }
        }
#else
        const int cur = 0;
        {
            const float* kp = K + base + (size_t)(j * 16) * AD;
            const float* vp = V + base + (size_t)(j * 16) * AD;
            #pragma unroll
            for (int u = 0; u < 4; ++u) {           // 128 threads x 4 float4 each
                const int idx = tid + u * 128;
                const int r   = idx >> 5;
                const int cc  = (idx & 31) * 4;
                *(v4f*)&Ks[0][r][cc] = *(const v4f*)(kp + r * AD + cc);
                *(v4f*)&Vs[0][r][cc] = *(const v4f*)(vp + r * AD + cc);
            }
        }
        if (j < qtmax) {
            const float* kn = K + base + (size_t)((j + 1) * 16) * AD;
            __builtin_prefetch(kn + tid * 16, 0, 0);
        }
#endif
        __syncthreads();

        if (j <= qt) {
            // ---- S = Q * K^T : B[e][n] = K[n][e] ----
            v8f s = {};
            #pragma unroll
            for (int e = 0; e < 32; ++e) {
                const int eb = e * 4 + 2 * hi;
                v2f bk = { Ks[cur][lm][eb], Ks[cur][lm][eb + 1] };
                s = wmma_f32(qa[e], bk, s);
            }

            // ---- online softmax (rows live in 16-lane halves) ----
            float alpha[8];
            #pragma unroll
            for (int i = 0; i < 8; ++i) {
                const int kidx = j * 16 + lm;
                const int qidx = qt * 16 + hi * 8 + i;
                float sv = (kidx <= qidx) ? s[i] * scale : -__builtin_inff();

                float r = sv;
                r = fmaxf(r, __shfl_xor(r, 1));
                r = fmaxf(r, __shfl_xor(r, 2));
                r = fmaxf(r, __shfl_xor(r, 4));
                r = fmaxf(r, __shfl_xor(r, 8));

                const float mnew = fmaxf(mrow[i], r);
                alpha[i] = (mrow[i] == -__builtin_inff())
                             ? 0.0f : __expf(mrow[i] - mnew);

                const float p = (sv == -__builtin_inff())
                             ? 0.0f : __expf(sv - mnew);
                float ps = p;
                ps += __shfl_xor(ps, 1);
                ps += __shfl_xor(ps, 2);
                ps += __shfl_xor(ps, 4);
                ps += __shfl_xor(ps, 8);

                lrow[i] = lrow[i] * alpha[i] + ps;
                mrow[i] = mnew;
                Ps[wv][hi * 8 + i][lm] = p;     // C-layout -> LDS (A-frag transpose)
            }
            #pragma unroll
            for (int t = 0; t < 8; ++t)
                #pragma unroll
                for (int i = 0; i < 8; ++i) o[t][i] *= alpha[i];
        }
        __syncthreads();

        if (j <= qt) {
            // ---- O += P * V : A = P (K-loop of 4), B = V tile rows ----
            #pragma unroll
            for (int kk = 0; kk < 4; ++kk) {
                const int kb = kk * 4 + 2 * hi;
                v2f pa = { Ps[wv][lm][kb], Ps[wv][lm][kb + 1] };
                #pragma unroll
                for (int t = 0; t < 8; ++t) {
                    v2f bv = { Vs[cur][kb][t * 16 + lm],
                               Vs[cur][kb + 1][t * 16 + lm] };
                    o[t] = wmma_f32(pa, bv, o[t]);
                }
            }
        }
        __syncthreads();        // all waves done with buf `cur` before next issue
    }

    // ---- normalize, round to 4 decimals (RTE matches jnp.round), store ----
    #pragma unroll
    for (int t = 0; t < 8; ++t) {
        #pragma unroll
        for (int i = 0; i < 8; ++i) {
            float val = o[t][i] / lrow[i];
            val = __builtin_rintf(val * 1.0e4f) * 1.0e-4f;
            O[base + (size_t)(qt * 16 + hi * 8 + i) * AD + t * 16 + lm] = val;
        }
    }
}

// ---------------------------------------------------------------------------
extern "C" void kernel_launch(void* const* d_in, const int* in_sizes, int n_in,
                              void* d_out, int out_size, void* d_ws, size_t ws_size,
                              hipStream_t stream) {
    const float* X  = (const float*)d_in[0];   // embedded [4,4096,1024]
    const float* Wk = (const float*)d_in[1];   // [128,1024]
    const float* Wq = (const float*)d_in[2];
    const float* Wv = (const float*)d_in[3];
    float* Out = (float*)d_out;                // [4,4096,128]

    const size_t nQKV = (size_t)NBAT * SEQ * AD;
    float* Qw = (float*)d_ws;
    float* Kw = Qw + nQKV;
    float* Vw = Kw + nQKV;

    qkv_proj_kernel<<<dim3(NBAT * SEQ / 16), 256, 0, stream>>>(
        X, Wq, Wk, Wv, Qw, Kw, Vw);
    attn_kernel<<<dim3(NBAT * (SEQ / 16) / AWAVES), 32 * AWAVES, 0, stream>>>(
        Qw, Kw, Vw, Out);
}